// InterFramePoseCalibrater_72902774882478
// MI455X (gfx1250) — compile-verified
//
#include <hip/hip_runtime.h>

namespace {
constexpr int Cc  = 32;
constexpr int Dd  = 64;
constexpr int Hh  = 192;
constexpr int Ww  = 192;
constexpr int DHW = Dd * Hh * Ww;
constexpr int WT  = 4;   // w-tile per block (blockDim.y)
}

// One-wave setup kernel: compose the whole coordinate transform into a 3x4
// affine over (h, w, d):  [ix, iy, iz] = A * [h, w, d]^T + b, stored as 12
// floats in workspace. Mirrors the reference math:
//   past = R_past^T * (R_cur * p + t_cur - t_past),  p = voxelsize*(q - center)
//   ix (input W axis) <- past_z, iy (H axis) <- past_x, iz (D axis) <- past_y
__global__ void pose_compose_kernel(const float* __restrict__ vs_p,
                                    const float* __restrict__ Rp,
                                    const float* __restrict__ tp,
                                    const float* __restrict__ Rc,
                                    const float* __restrict__ tc,
                                    float* __restrict__ ws) {
  if (threadIdx.x != 0) return;
  const float vs = vs_p[0];
  float M[3][3], cv[3];
  for (int i = 0; i < 3; ++i) {
    cv[i] = Rp[0 + i] * (tc[0] - tp[0]) +
            Rp[3 + i] * (tc[1] - tp[1]) +
            Rp[6 + i] * (tc[2] - tp[2]);
    for (int j = 0; j < 3; ++j) {
      M[i][j] = Rp[0 + i] * Rc[0 + j] +
                Rp[3 + i] * Rc[3 + j] +
                Rp[6 + i] * Rc[6 + j];
    }
  }
  const float cH = (Hh - 1) * 0.5f;
  const float cW = (Ww - 1) * 0.5f;
  const float cD = (Dd - 1) * 0.5f;
  // voxelsize cancels in the linear coefficients:
  const float sx = (float)Ww / (float)(Dd - 1);  // ix <- past_z (M row 2)
  const float sy = (float)Hh / (float)(Hh - 1);  // iy <- past_x (M row 0)
  const float sz = (float)Dd / (float)(Ww - 1);  // iz <- past_y (M row 1)
  float A[3][3], b[3];
  for (int j = 0; j < 3; ++j) {
    A[0][j] = sx * M[2][j];
    A[1][j] = sy * M[0][j];
    A[2][j] = sz * M[1][j];
  }
  b[0] = (float)Ww * cv[2] / (vs * (float)(Dd - 1)) + (Ww - 1) * 0.5f;
  b[1] = (float)Hh * cv[0] / (vs * (float)(Hh - 1)) + (Hh - 1) * 0.5f;
  b[2] = (float)Dd * cv[1] / (vs * (float)(Ww - 1)) + (Dd - 1) * 0.5f;
  for (int r = 0; r < 3; ++r) {
    b[r] -= A[r][0] * cH + A[r][1] * cW + A[r][2] * cD;  // fold centers into b
    ws[4 * r + 0] = A[r][0];
    ws[4 * r + 1] = A[r][1];
    ws[4 * r + 2] = A[r][2];
    ws[4 * r + 3] = b[r];
  }
}

// Main gather kernel. Lanes vary along output-d so the 8 corner gathers are
// coalesced along the input-W (contiguous) axis (ix ~ 3.05*d).
// blockDim = (64, 4) = 8 wave32s; each wave holds 32 consecutive d.
__global__ __launch_bounds__(256) void pose_warp_kernel(
    const float* __restrict__ in,   // [C, D, H, W]
    const float* __restrict__ ws,   // 12 floats: 3x(A_h, A_w, A_d, b)
    float* __restrict__ out) {      // [C, D, H, W]
  const int d  = threadIdx.x;                     // 0..63 (full D)
  const int wb = blockIdx.x % (Ww / WT);
  const int h  = blockIdx.x / (Ww / WT);
  const int w  = wb * WT + threadIdx.y;

  const float4 Ax = ((const float4*)ws)[0];
  const float4 Ay = ((const float4*)ws)[1];
  const float4 Az = ((const float4*)ws)[2];

  const float fh = (float)h, fw = (float)w, fd = (float)d;
  const float ix = fmaf(Ax.z, fd, fmaf(Ax.y, fw, fmaf(Ax.x, fh, Ax.w)));
  const float iy = fmaf(Ay.z, fd, fmaf(Ay.y, fw, fmaf(Ay.x, fh, Ay.w)));
  const float iz = fmaf(Az.z, fd, fmaf(Az.y, fw, fmaf(Az.x, fh, Az.w)));

  const float fx = floorf(ix), fy = floorf(iy), fz = floorf(iz);
  const float tx = ix - fx, ty = iy - fy, tz = iz - fz;
  const int x0 = (int)fx, y0 = (int)fy, z0 = (int)fz;
  const int x1 = x0 + 1, y1 = y0 + 1, z1 = z0 + 1;

  // padding_mode='zeros': per-axis validity folded into per-axis weights.
  const float wx0 = (x0 >= 0 && x0 < Ww) ? (1.0f - tx) : 0.0f;
  const float wx1 = (x1 >= 0 && x1 < Ww) ? tx : 0.0f;
  const float wy0 = (y0 >= 0 && y0 < Hh) ? (1.0f - ty) : 0.0f;
  const float wy1 = (y1 >= 0 && y1 < Hh) ? ty : 0.0f;
  const float wz0 = (z0 >= 0 && z0 < Dd) ? (1.0f - tz) : 0.0f;
  const float wz1 = (z1 >= 0 && z1 < Dd) ? tz : 0.0f;

  const int xi0 = min(max(x0, 0), Ww - 1), xi1 = min(max(x1, 0), Ww - 1);
  const int yi0 = min(max(y0, 0), Hh - 1), yi1 = min(max(y1, 0), Hh - 1);
  const int zi0 = min(max(z0, 0), Dd - 1), zi1 = min(max(z1, 0), Dd - 1);

  const int r00 = (zi0 * Hh + yi0) * Ww;
  const int r01 = (zi0 * Hh + yi1) * Ww;
  const int r10 = (zi1 * Hh + yi0) * Ww;
  const int r11 = (zi1 * Hh + yi1) * Ww;
  const int o0 = r00 + xi0, o1 = r00 + xi1;
  const int o2 = r01 + xi0, o3 = r01 + xi1;
  const int o4 = r10 + xi0, o5 = r10 + xi1;
  const int o6 = r11 + xi0, o7 = r11 + xi1;

  const float w0 = wz0 * wy0 * wx0, w1 = wz0 * wy0 * wx1;
  const float w2 = wz0 * wy1 * wx0, w3 = wz0 * wy1 * wx1;
  const float w4 = wz1 * wy0 * wx0, w5 = wz1 * wy0 * wx1;
  const float w6 = wz1 * wy1 * wx0, w7 = wz1 * wy1 * wx1;

  const int sidx = (d * Hh + h) * Ww + w;

  // 8-deep unroll: up to 64 independent global_load_b32 in flight per thread.
  // Channel base (c*DHW) is wave-uniform -> SGPR saddr, per-lane 32-bit
  // voffset reused across all 32 channels.
#pragma unroll 8
  for (int c = 0; c < Cc; ++c) {
    const float* bp = in + (size_t)c * DHW;
    const float v0 = bp[o0];
    const float v1 = bp[o1];
    const float v2 = bp[o2];
    const float v3 = bp[o3];
    const float v4 = bp[o4];
    const float v5 = bp[o5];
    const float v6 = bp[o6];
    const float v7 = bp[o7];
    float acc = v0 * w0;
    acc = fmaf(v1, w1, acc);
    acc = fmaf(v2, w2, acc);
    acc = fmaf(v3, w3, acc);
    acc = fmaf(v4, w4, acc);
    acc = fmaf(v5, w5, acc);
    acc = fmaf(v6, w6, acc);
    acc = fmaf(v7, w7, acc);
    // Write-once output: non-temporal store (TH=NT) keeps L2 for the gather
    // working set, which carries the 8x corner reuse.
    __builtin_nontemporal_store(acc, out + (size_t)c * DHW + sidx);
  }
}

extern "C" void kernel_launch(void* const* d_in, const int* in_sizes, int n_in,
                              void* d_out, int out_size, void* d_ws, size_t ws_size,
                              hipStream_t stream) {
  const float* past_grid  = (const float*)d_in[0];
  const float* voxelsize  = (const float*)d_in[1];
  const float* rot_past   = (const float*)d_in[2];
  const float* trans_past = (const float*)d_in[3];
  const float* rot_cur    = (const float*)d_in[4];
  const float* trans_cur  = (const float*)d_in[5];
  float* ws  = (float*)d_ws;
  float* out = (float*)d_out;

  pose_compose_kernel<<<1, 32, 0, stream>>>(voxelsize, rot_past, trans_past,
                                            rot_cur, trans_cur, ws);

  dim3 block(Dd, WT);                 // (64, 4) = 256 threads = 8 wave32s
  dim3 grid(Hh * (Ww / WT));          // 192 * 48 = 9216 blocks
  pose_warp_kernel<<<grid, block, 0, stream>>>(past_grid, ws, out);
}